// GNNConv_15126874817103
// MI455X (gfx1250) — compile-verified
//
#include <hip/hip_runtime.h>

typedef __attribute__((ext_vector_type(2))) float v2f;
typedef __attribute__((ext_vector_type(8))) float v8f;

#define N_NODES 50000
#define N_EDGES 640000
#define DIM     128
#define NND     (N_NODES * DIM)          // 6,400,000
#define LDSP    132                      // padded pitch: 132 % 64 = 4 -> conflict-free row access

// Non-returning FP32 atomic add straight to the L2 atomic units (STOREcnt path).
__device__ __forceinline__ void atom_add_f32(float* p, float v) {
    asm volatile("global_atomic_add_f32 %0, %1, off" :: "v"(p), "v"(v) : "memory");
}

// CDNA5 async memory->LDS copy (ASYNCcnt path, no VGPR round-trip).
// ldsOff: LDS byte address (low 32 bits of the flat shared address).
__device__ __forceinline__ void async_load_b128(unsigned ldsOff, const float* gaddr) {
    asm volatile("global_load_async_to_lds_b128 %0, %1, off"
                 :: "v"(ldsOff), "v"(gaddr) : "memory");
}
__device__ __forceinline__ void wait_async0() {
    asm volatile("s_wait_asynccnt 0x0" ::: "memory");
}

// One 16x16 f32 output tile: 32 chained V_WMMA_F32_16X16X4_F32 over K=128.
// ldsRowA points at this lane's A-row (row = lane%16) in LDS; wrow is W[n]*, n = this lane's column.
__device__ __forceinline__ v8f wmma_k128(const float* ldsRowA, const float* wrow,
                                         int khalf, v8f acc) {
#pragma unroll
    for (int k = 0; k < 32; ++k) {
        const int kb = 4 * k + 2 * khalf;          // lanes 0-15: K=kb,kb+1 ; lanes 16-31: +2
        v2f a = *(const v2f*)(ldsRowA + kb);
        v2f b = *(const v2f*)(wrow + kb);          // B[k][n] = W[n][k]
        acc = __builtin_amdgcn_wmma_f32_16x16x4_f32(false, a, false, b,
                                                    (short)0, acc, false, false);
    }
    return acc;
}

// ---------------------------------------------------------------------------
// Kernel 1: fused node GEMMs.  Block = 128 threads (4 waves). Each block owns a
// 16-row tile of x; wave w computes one of {A,B,D,E}: Y = x @ W^T + b.
// ---------------------------------------------------------------------------
__global__ __launch_bounds__(128) void node_gemm_kernel(
    const float* __restrict__ x,
    const float* __restrict__ Aw, const float* __restrict__ Ab,
    const float* __restrict__ Bw, const float* __restrict__ Bb,
    const float* __restrict__ Dw, const float* __restrict__ Db,
    const float* __restrict__ Ew, const float* __restrict__ Eb,
    float* __restrict__ Axo, float* __restrict__ Bxo,
    float* __restrict__ Dxo, float* __restrict__ Exo)
{
    __shared__ float ldsX[16 * LDSP];
    const int  tid   = threadIdx.x;
    const int  wave  = tid >> 5;
    const int  lane  = tid & 31;
    const int  lcol  = lane & 15;
    const int  khalf = lane >> 4;
    const long rowBase = (long)blockIdx.x * 16;

    { // stage 16x128 x-tile, 16B-vector loads/stores
        const int r = tid >> 3;
        const int c = (tid & 7) * 16;
        const float4* s = (const float4*)(x + (rowBase + r) * DIM + c);
        float4*       d = (float4*)(ldsX + r * LDSP + c);
        d[0] = s[0]; d[1] = s[1]; d[2] = s[2]; d[3] = s[3];
    }
    __syncthreads();

    const float* W   = (wave == 0) ? Aw : (wave == 1) ? Bw : (wave == 2) ? Dw : Ew;
    const float* bia = (wave == 0) ? Ab : (wave == 1) ? Bb : (wave == 2) ? Db : Eb;
    float*       Y   = (wave == 0) ? Axo : (wave == 1) ? Bxo : (wave == 2) ? Dxo : Exo;

    const float* ldsRow = ldsX + lcol * LDSP;
    for (int nt = 0; nt < 8; ++nt) {
        const int n = nt * 16 + lcol;
        const float bv = bia[n];
        v8f acc;
#pragma unroll
        for (int i = 0; i < 8; ++i) acc[i] = bv;   // C(m,n) = bias[n]
        acc = wmma_k128(ldsRow, W + (long)n * DIM, khalf, acc);
#pragma unroll
        for (int v = 0; v < 8; ++v) {
            const long m = v + 8 * khalf;          // D row per VGPR
            Y[(rowBase + m) * DIM + n] = acc[v];
        }
    }
}

// ---------------------------------------------------------------------------
// Kernel 2: fused edge pipeline. Block = 256 (8 waves); each wave owns one
// 16-edge tile: async-stage e tile to LDS, Ce via WMMA, gather Dx[dst]+Ex[src],
// sigmoid, scatter atomics, edge BN+ReLU+residual (NT stores).
// ---------------------------------------------------------------------------
__global__ __launch_bounds__(256) void edge_kernel(
    const float* __restrict__ e,
    const long long* __restrict__ srcIdx, const long long* __restrict__ dstIdx,
    const float* __restrict__ Cw, const float* __restrict__ Cb,
    const float* __restrict__ beg, const float* __restrict__ beb,
    const float* __restrict__ bem, const float* __restrict__ bev,
    const float* __restrict__ Dx, const float* __restrict__ Ex,
    const float* __restrict__ Bx,
    float* __restrict__ num, float* __restrict__ den,
    float* __restrict__ e_out)
{
    __shared__ float ldsE[8][16 * LDSP];
    const int  tid   = threadIdx.x;
    const int  wave  = tid >> 5;
    const int  lane  = tid & 31;
    const int  lcol  = lane & 15;
    const int  khalf = lane >> 4;
    const long eBase = ((long)blockIdx.x * 8 + wave) * 16;
    float* myLds = &ldsE[wave][0];

    // Warm the (L2-resident, block-shared) weight: global_prefetch_b8.
    __builtin_prefetch(Cw + tid * 64, 0, 1);

    { // async-stage this wave's 16x128 e-tile: row j, lane -> 16B chunk
        const unsigned ldsBase = (unsigned)(unsigned long long)myLds;
#pragma unroll
        for (int j = 0; j < 16; ++j) {
            const unsigned ldsOff = ldsBase + (unsigned)(j * LDSP + lane * 4) * 4u;
            async_load_b128(ldsOff, e + (eBase + j) * DIM + lane * 4);
        }
    }

    long sN[8], dN[8];
#pragma unroll
    for (int v = 0; v < 8; ++v) {
        const long m = v + 8 * khalf;              // rows this lane owns in C/D
        sN[v] = (long)srcIdx[eBase + m];
        dN[v] = (long)dstIdx[eBase + m];
    }

    wait_async0();
    __syncthreads();

    const float* ldsRow = myLds + lcol * LDSP;
    for (int nt = 0; nt < 8; ++nt) {
        const int n   = nt * 16 + lcol;
        const float bv  = Cb[n];
        const float gm  = beg[n], bt = beb[n], mn = bem[n];
        const float inv = rsqrtf(bev[n] + 1e-5f);
        v8f acc;
#pragma unroll
        for (int i = 0; i < 8; ++i) acc[i] = bv;
        acc = wmma_k128(ldsRow, Cw + (long)n * DIM, khalf, acc);   // Ce tile
#pragma unroll
        for (int v = 0; v < 8; ++v) {
            const long m = v + 8 * khalf;
            const long s = sN[v], d = dN[v];
            const float eij = acc[v] + Dx[d * DIM + n] + Ex[s * DIM + n];
            const float sig = 1.0f / (1.0f + __expf(-eij));
            atom_add_f32(num + d * DIM + n, sig * Bx[s * DIM + n]);
            atom_add_f32(den + d * DIM + n, sig);
            float r = (eij - mn) * inv * gm + bt;
            r = fmaxf(r, 0.0f);
            // streamed-once output: non-temporal store to spare L2 for gathers
            __builtin_nontemporal_store(myLds[m * LDSP + n] + r,
                                        e_out + (eBase + m) * DIM + n);
        }
    }
}

// ---------------------------------------------------------------------------
// Kernel 3: node finalize:  x_out = x + relu(BN(Ax + num/(den+eps)))
// ---------------------------------------------------------------------------
__global__ __launch_bounds__(256) void node_final_kernel(
    const float* __restrict__ x, const float* __restrict__ Axo,
    const float* __restrict__ num, const float* __restrict__ den,
    const float* __restrict__ bxg, const float* __restrict__ bxb,
    const float* __restrict__ bxm, const float* __restrict__ bxv,
    float* __restrict__ x_out)
{
    const long i = (long)blockIdx.x * blockDim.x + threadIdx.x;
    if (i >= (long)NND) return;
    const int c = (int)(i & (DIM - 1));
    const float aggr = num[i] / (den[i] + 1e-6f);
    const float xn   = Axo[i] + aggr;
    float r = (xn - bxm[c]) * rsqrtf(bxv[c] + 1e-5f) * bxg[c] + bxb[c];
    x_out[i] = x[i] + fmaxf(r, 0.0f);
}

__global__ __launch_bounds__(256) void zero_kernel(float4* __restrict__ p, long n4)
{
    const long i = (long)blockIdx.x * blockDim.x + threadIdx.x;
    if (i < n4) p[i] = make_float4(0.f, 0.f, 0.f, 0.f);
}

// ---------------------------------------------------------------------------
extern "C" void kernel_launch(void* const* d_in, const int* in_sizes, int n_in,
                              void* d_out, int out_size, void* d_ws, size_t ws_size,
                              hipStream_t stream)
{
    const float* x   = (const float*)d_in[0];
    const float* e   = (const float*)d_in[1];
    const float* Aw  = (const float*)d_in[2];
    const float* Ab  = (const float*)d_in[3];
    const float* Bw  = (const float*)d_in[4];
    const float* Bb  = (const float*)d_in[5];
    const float* Cw  = (const float*)d_in[6];
    const float* Cb  = (const float*)d_in[7];
    const float* Dw  = (const float*)d_in[8];
    const float* Db  = (const float*)d_in[9];
    const float* Ew  = (const float*)d_in[10];
    const float* Eb  = (const float*)d_in[11];
    const float* bxg = (const float*)d_in[12];
    const float* bxb = (const float*)d_in[13];
    const float* bxm = (const float*)d_in[14];
    const float* bxv = (const float*)d_in[15];
    const float* beg = (const float*)d_in[16];
    const float* beb = (const float*)d_in[17];
    const float* bem = (const float*)d_in[18];
    const float* bev = (const float*)d_in[19];
    const long long* ei  = (const long long*)d_in[20];
    const long long* srcI = ei;              // edge_index[0]
    const long long* dstI = ei + N_EDGES;    // edge_index[1]

    float* out   = (float*)d_out;
    float* x_out = out;
    float* e_out = out + (long)NND;

    float* ws  = (float*)d_ws;               // 6 * NND floats = 153.6 MB
    float* Axo = ws + 0L * NND;
    float* Bxo = ws + 1L * NND;
    float* Dxo = ws + 2L * NND;
    float* Exo = ws + 3L * NND;
    float* num = ws + 4L * NND;
    float* den = ws + 5L * NND;

    // 1) zero the scatter accumulators (num+den contiguous: 2*NND floats)
    {
        const long n4 = (2L * NND) / 4;       // 3,200,000 float4
        zero_kernel<<<(int)((n4 + 255) / 256), 256, 0, stream>>>((float4*)num, n4);
    }
    // 2) fused node GEMMs (Ax,Bx,Dx,Ex): 50000/16 = 3125 row-tiles
    node_gemm_kernel<<<N_NODES / 16, 128, 0, stream>>>(
        x, Aw, Ab, Bw, Bb, Dw, Db, Ew, Eb, Axo, Bxo, Dxo, Exo);
    // 3) fused edge pipeline: 640000/16 tiles / 8 waves-per-block = 5000 blocks
    edge_kernel<<<N_EDGES / (16 * 8), 256, 0, stream>>>(
        e, srcI, dstI, Cw, Cb, beg, beb, bem, bev,
        Dxo, Exo, Bxo, num, den, e_out);
    // 4) node finalize
    node_final_kernel<<<(NND + 255) / 256, 256, 0, stream>>>(
        x, Axo, num, den, bxg, bxb, bxm, bxv, x_out);
}